// HighOrderFactorizationMachineModel_86517821214551
// MI455X (gfx1250) — compile-verified
//
#include <hip/hip_runtime.h>
#include <cstdint>
#include <cstddef>

// HighOrderFactorizationMachine (B=4096, F=24, D=16, ORDER=3)
// y[b] = sigmoid( bias + sum_f w[idx] + e2(emb[idx][:16]) + e3(emb[idx][16:32]) )
//
// One wave32 per sample. 24 embedding rows x 128B are gathered with the CDNA5
// async DMA path (global_load_async_to_lds_b128, ASYNCcnt) into LDS, then each
// lane owns one of the 32 dims and runs the per-dim reductions.

#define F_FIELDS        24
#define EMB_DIM         32          // D * (ORDER - 1)
#define WAVES_PER_BLOCK 8
#define THREADS         (WAVES_PER_BLOCK * 32)

__global__ __launch_bounds__(THREADS)
void hofm_fwd_kernel(const int* __restrict__ x,
                     const int* __restrict__ offsets,
                     const float* __restrict__ w_linear,
                     const float* __restrict__ bias,
                     const float* __restrict__ emb,
                     float* __restrict__ out,
                     int nB)
{
    // 8 waves * 24 rows * 128B = 24 KB static LDS (well under 320 KB/WGP)
    __shared__ __align__(16) float smem[WAVES_PER_BLOCK * F_FIELDS * EMB_DIM];

    const int lane = threadIdx.x & 31;
    const int wid  = threadIdx.x >> 5;
    const int b    = blockIdx.x * WAVES_PER_BLOCK + wid;
    if (b >= nB) return;                       // per-wave uniform; no block barriers used

    // ---- per-lane field index + linear term (lanes 0..23) --------------------
    int   idxv  = 0;
    float wpart = 0.0f;
    if (lane < F_FIELDS) {
        idxv  = x[b * F_FIELDS + lane] + offsets[lane];
        wpart = w_linear[idxv];
    }

    // ---- async gather: 24 rows x 128B -> LDS (6 x b128, 4 rows per instr) ----
    // lane -> (row-in-group-of-4, 16B chunk): 8 lanes cover one 128B row.
    const unsigned ldsbase = (unsigned)(uintptr_t)(&smem[wid * (F_FIELDS * EMB_DIM)]);
    const int rsub = lane >> 3;            // 0..3
    const int csub = (lane & 7) << 4;      // byte offset 0,16,...,112
#pragma unroll
    for (int k = 0; k < 6; ++k) {
        const int row    = k * 4 + rsub;                   // 0..23
        const int rowIdx = __shfl(idxv, row);              // wave-uniform per row
        const char* gaddr = (const char*)emb
                          + ((size_t)(unsigned)rowIdx << 7)   // idx * 128B
                          + (size_t)csub;
        const unsigned laddr = ldsbase + (unsigned)((row << 7) + csub);
        asm volatile("global_load_async_to_lds_b128 %0, %1, off"
                     :: "v"(laddr), "v"(gaddr)
                     : "memory");
    }
    // wait for this wave's async DMA writes to land in LDS
    asm volatile("s_wait_asynccnt 0" ::: "memory");

    // ---- per-dim reductions: lane d walks column d (bank-conflict-free) ------
    const float* col = &smem[wid * (F_FIELDS * EMB_DIM) + lane];
    float s1 = 0.f, s2 = 0.f;              // FM part   (dims 0..15)
    float e1 = 0.f, e2 = 0.f, e3 = 0.f;    // ANOVA e3  (dims 16..31)
#pragma unroll
    for (int f = 0; f < F_FIELDS; ++f) {
        const float v = col[f * EMB_DIM];
        e3 = fmaf(e2, v, e3);              // update high orders first
        e2 = fmaf(e1, v, e2);
        e1 += v;
        s1 += v;
        s2 = fmaf(v, v, s2);
    }
    float contrib = (lane < 16) ? (0.5f * (s1 * s1 - s2)) : e3;
    contrib += wpart;

    // ---- wave32 butterfly reduction, then sigmoid ----------------------------
#pragma unroll
    for (int off = 16; off > 0; off >>= 1)
        contrib += __shfl_xor(contrib, off);

    if (lane == 0) {
        const float y = contrib + bias[0];
        out[b] = 1.0f / (1.0f + __expf(-y));
    }
}

extern "C" void kernel_launch(void* const* d_in, const int* in_sizes, int n_in,
                              void* d_out, int out_size, void* d_ws, size_t ws_size,
                              hipStream_t stream)
{
    const int*   x        = (const int*)  d_in[0];   // (B, 24) int32
    const int*   offsets  = (const int*)  d_in[1];   // (24,)   int32
    const float* w_linear = (const float*)d_in[2];   // (V, 1)  f32
    const float* bias     = (const float*)d_in[3];   // (1,)    f32
    const float* emb      = (const float*)d_in[4];   // (V, 32) f32
    float*       out      = (float*)      d_out;     // (B,)    f32

    const int nB     = in_sizes[0] / F_FIELDS;
    const int blocks = (nB + WAVES_PER_BLOCK - 1) / WAVES_PER_BLOCK;

    hipLaunchKernelGGL(hofm_fwd_kernel, dim3(blocks), dim3(THREADS), 0, stream,
                       x, offsets, w_linear, bias, emb, out, nB);
}